// KineticEquation_59304908423466
// MI455X (gfx1250) — compile-verified
//
#include <hip/hip_runtime.h>
#include <hip/hip_bf16.h>

#define NS      1024   // species
#define BATCH   512
#define ROWS    8      // batch rows per block tile
#define RSPLIT  4      // reaction chunks (blocks combining via global fp atomics)
#define TPB     256

typedef __attribute__((address_space(1))) int gint;
typedef __attribute__((address_space(3))) int sint;

__global__ __launch_bounds__(TPB) void kin_zero_kernel(float* __restrict__ out, int n) {
    int i = blockIdx.x * TPB + threadIdx.x;
    if (i < n) out[i] = 0.0f;
}

__global__ __launch_bounds__(TPB) void KineticEquation_kernel(
    const float* __restrict__ y_in,
    const int*   __restrict__ i1r, const int* __restrict__ i1p,
    const float* __restrict__ rate1, int R1,
    const int*   __restrict__ i2r0, const int* __restrict__ i2r1,
    const int*   __restrict__ i2p,
    const float* __restrict__ rate2, int R2,
    float* __restrict__ y_out)
{
    __shared__ float y_s[NS * ROWS];   // transposed tile: y_s[s*ROWS + row]
    __shared__ float acc[NS * ROWS];   // transposed accumulator

    const int t    = threadIdx.x;
    const int wave = t >> 5;           // 0..7, doubles as "row" for stage/writeback
    const int lane = t & 31;
    const int r0   = blockIdx.x * ROWS;    // batch-row tile base
    const int chunk = blockIdx.y;          // reaction chunk

    // Zero the LDS accumulator.
    for (int i = t; i < NS * ROWS; i += TPB) acc[i] = 0.0f;

    // ---- Stage y tile into LDS, transposed. Wave w stages batch row w. ----
    {
        const int row = wave;
        const float* src = y_in + (size_t)(r0 + row) * NS;
#if __has_builtin(__builtin_amdgcn_global_load_async_to_lds_b32)
        // CDNA5 async global->LDS: per-lane LDS destination performs the transpose.
        float* srcnc = const_cast<float*>(src);
        gint* gsrc = (gint*)srcnc;
        sint* ldst = (sint*)&y_s[0];
        for (int s = lane; s < NS; s += 32) {
            __builtin_amdgcn_global_load_async_to_lds_b32(gsrc + s,
                                                          ldst + s * ROWS + row,
                                                          0, 0);
        }
#if __has_builtin(__builtin_amdgcn_s_wait_asynccnt)
        __builtin_amdgcn_s_wait_asynccnt(0);
#else
        asm volatile("s_wait_asynccnt 0" ::: "memory");
#endif
#else
        for (int s = lane; s < NS; s += 32)
            y_s[s * ROWS + row] = src[s];
#endif
    }
    __syncthreads();

    // ---- Reaction processing: group of 8 lanes shares one reaction, lane%8 = row. ----
    {
        const int row = t & (ROWS - 1);
        const int grp = t >> 3;            // 0..31 reaction slots per block

        const int r1_per = (R1 + RSPLIT - 1) / RSPLIT;
        const int r1_lo  = chunk * r1_per;
        const int r1_hi  = (r1_lo + r1_per < R1) ? (r1_lo + r1_per) : R1;
        for (int r = r1_lo + grp; r < r1_hi; r += (TPB / ROWS)) {
            const int   ir = i1r[r];
            const int   ip = i1p[r];
            const float v  = y_s[ir * ROWS + row] * rate1[r];
            atomicAdd(&acc[ip * ROWS + row], v);       // ds_add_f32 (no return)
        }

        const int r2_per = (R2 + RSPLIT - 1) / RSPLIT;
        const int r2_lo  = chunk * r2_per;
        const int r2_hi  = (r2_lo + r2_per < R2) ? (r2_lo + r2_per) : R2;
        for (int r = r2_lo + grp; r < r2_hi; r += (TPB / ROWS)) {
            const int   a = i2r0[r];
            const int   b = i2r1[r];
            const int   p = i2p[r];
            const float v = y_s[a * ROWS + row] * y_s[b * ROWS + row] * rate2[r];
            atomicAdd(&acc[p * ROWS + row], v);        // ds_add_f32 (no return)
        }
    }
    __syncthreads();

    // ---- Writeback: wave w handles row w; coalesced global fp atomics. ----
    {
        const int row = wave;
        float* dst = y_out + (size_t)(r0 + row) * NS;
        for (int s = lane; s < NS; s += 32) {
            unsafeAtomicAdd(&dst[s], acc[s * ROWS + row]);  // global_atomic_add_f32
        }
    }
}

extern "C" void kernel_launch(void* const* d_in, const int* in_sizes, int n_in,
                              void* d_out, int out_size, void* d_ws, size_t ws_size,
                              hipStream_t stream) {
    (void)n_in; (void)d_ws; (void)ws_size;
    // setup_inputs order:
    // 0:t_in 1:y_in 2:inds_1r 3:inds_1p 4:rate_1 5:inds_2r0 6:inds_2r1 7:inds_2p 8:rate_2
    const float* y_in  = (const float*)d_in[1];
    const int*   i1r   = (const int*)  d_in[2];
    const int*   i1p   = (const int*)  d_in[3];
    const float* rate1 = (const float*)d_in[4];
    const int*   i2r0  = (const int*)  d_in[5];
    const int*   i2r1  = (const int*)  d_in[6];
    const int*   i2p   = (const int*)  d_in[7];
    const float* rate2 = (const float*)d_in[8];
    const int R1 = in_sizes[2];
    const int R2 = in_sizes[5];
    float* out = (float*)d_out;

    // Output is combined with atomics across reaction chunks -> must start at zero.
    kin_zero_kernel<<<(out_size + TPB - 1) / TPB, TPB, 0, stream>>>(out, out_size);

    dim3 grid(BATCH / ROWS, RSPLIT);
    KineticEquation_kernel<<<grid, TPB, 0, stream>>>(
        y_in, i1r, i1p, rate1, R1, i2r0, i2r1, i2p, rate2, R2, out);
}